// DCF_89653147336821
// MI455X (gfx1250) — compile-verified
//
#include <hip/hip_runtime.h>
#include <math.h>

typedef __attribute__((ext_vector_type(2))) float v2f;
typedef __attribute__((ext_vector_type(8))) float v8f;

#define ALPHA_C 0.35f
#define OMA     0.65f

// strides (floats)
// input  x[t, ri, b, c]  : t=4112, ri=2056, b=8, c=1   (8000,2,257,8)
// null_w w[n, ri, b, c]  : n=4112, ri=2056, b=8, c=1   (beam 0 slice)
//
// workspace layout (float offsets):
//   cr  [8000][257][5]  @ 0
//   ci  [8000][257][5]  @ 10,280,000
//   pw  [8000][257]     @ 20,560,000
//   p0r [4][257][5]     @ 22,616,000
//   p0i [4][257][5]     @ 22,621,140      total 22,626,280 floats (~90.5 MB)
#define WS_CR  ((size_t)0)
#define WS_CI  ((size_t)10280000)
#define WS_PW  ((size_t)20560000)
#define WS_P0R ((size_t)22616000)
#define WS_P0I ((size_t)22621140)

// output: dcf (4,2000,257,5) @ 0 ; targ (4,2000,2,257) @ 10,280,000
#define TARG_OFF ((size_t)10280000)

// ---------------- Phase 1: WMMA beamforming + cr/ci/pw/targ ----------------
// grid = (257 bins, 125 time-groups), block = 128 (4 waves); each wave owns a
// 16-frame x 1-bin tile.  y(16x6, complex) = X(16x8, complex) * W^T via
// 8x V_WMMA_F32_16X16X4_F32.
__global__ __launch_bounds__(128) void dcf_bf_wmma(
    const float* __restrict__ x, const float* __restrict__ w,
    float* __restrict__ out, float* __restrict__ ws)
{
  const int b    = blockIdx.x;             // bin
  const int lane = threadIdx.x & 31;
  const int wave = threadIdx.x >> 5;
  const int t0   = blockIdx.y * 64 + wave * 16;   // 2000 % 16 == 0 -> no block straddle
  const int n    = lane & 15;              // B column / A row
  const int kh   = lane >> 4;              // K-half selector

  // B operands: W for this bin.  B[k,n] with k = 2*kh + {0,1} (channels c0+2kh..)
  v2f BrA = {0.f, 0.f}, BrB = {0.f, 0.f}, BiA = {0.f, 0.f}, BiB = {0.f, 0.f};
  if (n < 6) {
    const float* wb = w + (size_t)n * 4112 + (size_t)b * 8 + 2 * kh;
    BrA = *(const v2f*)(wb);               // wr, channels 0..3 slice
    BrB = *(const v2f*)(wb + 4);           // wr, channels 4..7 slice
    BiA = *(const v2f*)(wb + 2056);        // wi, channels 0..3 slice
    BiB = *(const v2f*)(wb + 2056 + 4);    // wi, channels 4..7 slice
  }
  v2f nBiA = -BiA, nBiB = -BiB;            // yi uses -Wi (no A/B neg on f32 WMMA)

  // A operands: X tile.  Row m = lane&15, K = channels (2*kh + {0,1}).
  const float* xb = x + (size_t)(t0 + n) * 4112 + (size_t)b * 8 + 2 * kh;
  v2f ArA = *(const v2f*)(xb);
  v2f ArB = *(const v2f*)(xb + 4);
  v2f AiA = *(const v2f*)(xb + 2056);
  v2f AiB = *(const v2f*)(xb + 2056 + 4);

  // yr = Xr*Wr^T + Xi*Wi^T ; yi = Xi*Wr^T - Xr*Wi^T
  v8f yr = {}, yi = {};
  yr = __builtin_amdgcn_wmma_f32_16x16x4_f32(false, ArA, false, BrA,  (short)0, yr, false, false);
  yr = __builtin_amdgcn_wmma_f32_16x16x4_f32(false, ArB, false, BrB,  (short)0, yr, false, false);
  yr = __builtin_amdgcn_wmma_f32_16x16x4_f32(false, AiA, false, BiA,  (short)0, yr, false, false);
  yr = __builtin_amdgcn_wmma_f32_16x16x4_f32(false, AiB, false, BiB,  (short)0, yr, false, false);
  yi = __builtin_amdgcn_wmma_f32_16x16x4_f32(false, AiA, false, BrA,  (short)0, yi, false, false);
  yi = __builtin_amdgcn_wmma_f32_16x16x4_f32(false, AiB, false, BrB,  (short)0, yi, false, false);
  yi = __builtin_amdgcn_wmma_f32_16x16x4_f32(false, ArA, false, nBiA, (short)0, yi, false, false);
  yi = __builtin_amdgcn_wmma_f32_16x16x4_f32(false, ArB, false, nBiB, (short)0, yi, false, false);

  // pw[t,b] = mean_c(xr^2 + xi^2): lane kh=0 holds ch {0,1,4,5}, kh=1 holds {2,3,6,7}
  float s = ArA.x*ArA.x + ArA.y*ArA.y + ArB.x*ArB.x + ArB.y*ArB.y
          + AiA.x*AiA.x + AiA.y*AiA.y + AiB.x*AiB.x + AiB.y*AiB.y;
  s += __shfl_xor(s, 16, 32);
  if (kh == 0)
    ws[WS_PW + (size_t)(t0 + n) * 257 + b] = s * 0.125f;

  // Extract C (m = r + 8*kh per lane-half, n = lane&15): cr/ci, targ, phi0
#pragma unroll
  for (int r = 0; r < 8; ++r) {
    float vr = yr[r], vi = yi[r];
    float tr = __shfl(vr, lane & 16, 32);  // targ (n==0) broadcast within half
    float ti = __shfl(vi, lane & 16, 32);
    const int t = t0 + r + 8 * kh;
    if (n >= 1 && n <= 5) {
      float crv = tr * vr + ti * vi;
      float civ = ti * vr - tr * vi;
      size_t idx = ((size_t)t * 257 + b) * 5 + (n - 1);
      ws[WS_CR + idx] = crv;
      ws[WS_CI + idx] = civ;
      if ((t % 2000) == 0) {               // block-start quirk: (1-a) on first product only
        int blk = t / 2000;
        size_t pidx = ((size_t)blk * 257 + b) * 5 + (n - 1);
        ws[WS_P0R + pidx] = OMA * tr * vr + ti * vi;
        ws[WS_P0I + pidx] = OMA * ti * vr - tr * vi;
      }
    } else if (n == 0) {                   // targ output (real, imag)
      out[TARG_OFF + (size_t)t * 514 + b]       = vr;
      out[TARG_OFF + (size_t)t * 514 + 257 + b] = vi;
    }
  }
}

// ---------------- Phase 2: per-block sequential IIR + bin reductions -------
// grid = 4 (one WG per block), 512 threads.  Elements = 257 bins x 5 nulls.
__global__ __launch_bounds__(512) void dcf_recur(
    const float* __restrict__ ws, float* __restrict__ out)
{
  const int blk = blockIdx.x;
  const int e   = threadIdx.x;

  __shared__ float psd_s[257];
  __shared__ float phi_s[1285];
  __shared__ float red_s[6];               // aft[0..4], pre

  const int nelem = (e < 261) ? 3 : 2;     // elements e, e+512, e+1024
  float phir[3], phii[3], phiv[3], psdv[3];

  // f = 0: phi0 from quirk arrays, psd0 = (1-a)*pw, dcf0 without ratio
  for (int j = 0; j < nelem; ++j) {
    int ei = e + j * 512;
    int bin = ei / 5, nn = ei % 5;
    size_t pidx = ((size_t)blk * 257 + bin) * 5 + nn;
    phir[j] = ws[WS_P0R + pidx];
    phii[j] = ws[WS_P0I + pidx];
  }
  if (e < 257)
    psd_s[e] = OMA * ws[WS_PW + (size_t)(blk * 2000) * 257 + e];
  __syncthreads();
  for (int j = 0; j < nelem; ++j) {
    int ei = e + j * 512;
    int bin = ei / 5, nn = ei % 5;
    float phi = sqrtf(phir[j] * phir[j] + phii[j] * phii[j]);
    float d = fminf(fmaxf(phi / psd_s[bin], 0.01f), 1.0f);
    out[((size_t)(blk * 2000) * 257 + bin) * 5 + nn] = d;
  }
  __syncthreads();                          // order dcf0 psd reads vs f=1 psd writes

  for (int f = 1; f < 2000; ++f) {
    const size_t tbase = (size_t)(blk * 2000 + f) * 257;
    for (int j = 0; j < nelem; ++j) {
      int ei = e + j * 512;
      int bin = ei / 5, nn = ei % 5;
      size_t idx = (tbase + bin) * 5 + nn;
      float crv = ws[WS_CR + idx];
      float civ = ws[WS_CI + idx];
      phir[j] = ALPHA_C * phir[j] + OMA * crv;
      phii[j] = ALPHA_C * phii[j] + OMA * civ;
      phiv[j] = sqrtf(phir[j] * phir[j] + phii[j] * phii[j]);
      phi_s[ei] = phiv[j];
    }
    if (e < 257)
      psd_s[e] = ALPHA_C * psd_s[e] + OMA * ws[WS_PW + tbase + e];
    __syncthreads();

    for (int j = 0; j < nelem; ++j)         // snapshot psd before next-iter overwrite
      psdv[j] = psd_s[(e + j * 512) / 5];
    if (e < 5) {                            // aft[n] = sum_{b=5..69} phi[b,n]
      float a = 0.f;
      for (int bb = 5; bb < 70; ++bb) a += phi_s[bb * 5 + e];
      red_s[e] = a;
    } else if (e == 5) {                    // pre = sum_{b=5..69} psd[b]
      float p = 0.f;
      for (int bb = 5; bb < 70; ++bb) p += psd_s[bb];
      red_s[5] = p;
    }
    __syncthreads();

    float pre = red_s[5];
    for (int j = 0; j < nelem; ++j) {
      int ei = e + j * 512;
      int bin = ei / 5, nn = ei % 5;
      float ratio = fminf(fmaxf(red_s[nn] / (pre + 1e-10f), 0.01f), 1.0f);
      float d = fminf(fmaxf(phiv[j] / psdv[j], 0.01f), 1.0f);
      out[(tbase + bin) * 5 + nn] = sqrtf(d * ratio);
    }
    // no trailing barrier needed: next-iter shared writes are ordered behind
    // this iteration's second barrier for every consumer of phi_s/psd_s/red_s
  }
}

extern "C" void kernel_launch(void* const* d_in, const int* in_sizes, int n_in,
                              void* d_out, int out_size, void* d_ws, size_t ws_size,
                              hipStream_t stream) {
  (void)in_sizes; (void)n_in; (void)out_size; (void)ws_size;
  const float* x  = (const float*)d_in[0];
  const float* nw = (const float*)d_in[1];   // BEAM_ID = 0 -> beam slice at offset 0
  float* out = (float*)d_out;
  float* ws  = (float*)d_ws;

  dim3 g1(257, 125), b1(128);
  hipLaunchKernelGGL(dcf_bf_wmma, g1, b1, 0, stream, x, nw, out, ws);
  hipLaunchKernelGGL(dcf_recur, dim3(4), dim3(512), 0, stream, ws, out);
}